// GCN_4449586118681
// MI455X (gfx1250) — compile-verified
//
#include <hip/hip_runtime.h>
#include <math.h>

typedef __attribute__((ext_vector_type(2))) float v2f;
typedef __attribute__((ext_vector_type(8))) float v8f;

#define SLOTS 49
#define NPB 256   // nodes per block in the reduction kernel

// ---------------------------------------------------------------------------
// Workspace layout (floats): deg[N] | dinv[N] | wsum[N] | aggX[N*49] | acc[64*16]
// ---------------------------------------------------------------------------

__global__ void gcn_init_ws(float* deg, float* wsum, float* aggX, float* acc, int N) {
    size_t i = (size_t)blockIdx.x * blockDim.x + threadIdx.x;
    size_t tot = (size_t)N * SLOTS;
    if (i < (size_t)N) { deg[i] = 1.0f; wsum[i] = 0.0f; }   // deg starts at 1 (self-loop)
    if (i < tot)       aggX[i] = 0.0f;
    if (i < 1024)      acc[i]  = 0.0f;                      // padded 64x16, rows 49..63 stay 0
}

__global__ void gcn_degree(const int* __restrict__ dstI, float* deg, int E) {
    int e = blockIdx.x * blockDim.x + threadIdx.x;
    if (e < E) atomicAdd(&deg[dstI[e]], 1.0f);
}

__global__ void gcn_dinv(const float* __restrict__ deg, float* dinv, int N) {
    int n = blockIdx.x * blockDim.x + threadIdx.x;
    if (n < N) dinv[n] = rsqrtf(deg[n]);   // deg >= 1 always (self-loops)
}

// One wave32 per edge: gather x[src,0:49], scale by norm, scatter-add to aggX[dst].
__global__ void gcn_edge(const int* __restrict__ srcI, const int* __restrict__ dstI,
                         const float* __restrict__ x, const float* __restrict__ dinv,
                         float* wsum, float* aggX, int E) {
    int lane = threadIdx.x & 31;
    int wave = (int)(((size_t)blockIdx.x * blockDim.x + threadIdx.x) >> 5);
    if (wave >= E) return;
    int s = srcI[wave];
    int d = dstI[wave];
    float nrm = dinv[s] * dinv[d];
    if (lane == 0) atomicAdd(&wsum[s], nrm);
    const float* xs = x    + (size_t)s * SLOTS;
    float*       ag = aggX + (size_t)d * SLOTS;
    atomicAdd(&ag[lane], xs[lane] * nrm);          // slots 0..31
    int l2 = lane + 32;
    if (l2 < SLOTS) atomicAdd(&ag[l2], xs[l2] * nrm);  // slots 32..48
}

// acc[s,c] = sum_v (wsum[v]+dinv[v]^2) * relu((aggX[v,s] + x[v,s]*dinv[v]^2)*W1[c] + b1[c])
// Thread t < 784 owns one (s,c); block reduces a chunk of NPB nodes, then one atomicAdd.
__global__ void __launch_bounds__(800)
gcn_node_reduce(const float* __restrict__ x, const float* __restrict__ dinv,
                const float* __restrict__ wsum, const float* __restrict__ aggX,
                const float* __restrict__ W1, const float* __restrict__ b1,
                float* acc, int N) {
    int t = threadIdx.x;
    if (t >= SLOTS * 16) return;
    int s = t >> 4;
    int c = t & 15;
    float w1c = W1[c];
    float b1c = b1[c];
    float a = 0.0f;
    int n0 = blockIdx.x * NPB;
    int n1 = n0 + NPB; if (n1 > N) n1 = N;
    for (int n = n0; n < n1; ++n) {
        float dv = dinv[n];
        float d2 = dv * dv;
        float w  = wsum[n] + d2;                                   // self-loop norm
        float ax = aggX[(size_t)n * SLOTS + s] + x[(size_t)n * SLOTS + s] * d2;
        float h  = ax * w1c + b1c;
        h = h > 0.0f ? h : 0.0f;                                   // ReLU
        a += w * h;
    }
    atomicAdd(&acc[s * 16 + c], a);
}

// Single wave32: P = acc[64,16] @ W2[16,32] via V_WMMA_F32_16X16X4_F32
// (4 M-tiles x 2 N-tiles x 4 K-steps), then scalar = sum_{s<49,d} (P/N + b2[d])*Wfc,
// out = sigmoid((scalar + bfc)/28).
__global__ void __launch_bounds__(32)
gcn_final(const float* __restrict__ acc, const float* __restrict__ W2,
          const float* __restrict__ b2, const float* __restrict__ Wfc,
          const float* __restrict__ bfc, float* out, float invN) {
    int lane = threadIdx.x;
    int m  = lane & 15;        // row (A) / col (B,C) within tile
    int hi = lane >> 4;        // half-wave selector

    v8f cacc[4][2] = {};       // 8 f32 accumulators x 8 tiles = 64 VGPRs

    for (int mt = 0; mt < 4; ++mt) {
        int row = mt * 16 + m;                       // rows 49..63 of acc are zero-padded
        for (int k0 = 0; k0 < 16; k0 += 4) {
            // 32-bit A 16x4 layout: lanes0-15 {K=k0,k0+1}, lanes16-31 {K=k0+2,k0+3}
            v2f a;
            a.x = acc[row * 16 + k0 + 2 * hi + 0];
            a.y = acc[row * 16 + k0 + 2 * hi + 1];
            for (int nt = 0; nt < 2; ++nt) {
                // B 4x16 layout (row striped across lanes): lanes0-15 rows {K,K+1},
                // lanes16-31 rows {K+2,K+3}; N = lane&15.
                v2f b;
                b.x = W2[(k0 + 2 * hi + 0) * 32 + nt * 16 + m];
                b.y = W2[(k0 + 2 * hi + 1) * 32 + nt * 16 + m];
                cacc[mt][nt] = __builtin_amdgcn_wmma_f32_16x16x4_f32(
                    /*neg_a=*/false, a, /*neg_b=*/false, b,
                    /*c_mod=*/(short)0, cacc[mt][nt],
                    /*reuse_a=*/false, /*reuse_b=*/false);
            }
        }
    }

    // C/D layout: VGPR i on lane L holds element (M = i + 8*(L>>4), N = L&15)
    float partial = 0.0f;
    for (int mt = 0; mt < 4; ++mt) {
        for (int nt = 0; nt < 2; ++nt) {
            for (int i = 0; i < 8; ++i) {
                int s = mt * 16 + i + 8 * hi;
                int d = nt * 16 + m;
                if (s < SLOTS) {
                    float pooled = cacc[mt][nt][i] * invN + b2[d];
                    partial += pooled * Wfc[s * 32 + d];
                }
            }
        }
    }
    for (int off = 16; off > 0; off >>= 1)
        partial += __shfl_down(partial, off, 32);
    if (lane == 0) {
        float z = (partial + bfc[0]) * (1.0f / 28.0f);
        out[0] = 1.0f / (1.0f + expf(-z));
    }
}

extern "C" void kernel_launch(void* const* d_in, const int* in_sizes, int n_in,
                              void* d_out, int out_size, void* d_ws, size_t ws_size,
                              hipStream_t stream) {
    const float* x   = (const float*)d_in[0];
    const int*   ei  = (const int*)  d_in[1];
    const float* W1  = (const float*)d_in[2];
    const float* b1  = (const float*)d_in[3];
    const float* W2  = (const float*)d_in[4];
    const float* b2  = (const float*)d_in[5];
    const float* Wfc = (const float*)d_in[6];
    const float* bfc = (const float*)d_in[7];

    int N = in_sizes[0] / SLOTS;   // 30000
    int E = in_sizes[1] / 2;       // 300000
    const int* srcI = ei;          // edge_index row 0
    const int* dstI = ei + E;      // edge_index row 1

    float* ws   = (float*)d_ws;
    float* deg  = ws;
    float* dinv = ws + (size_t)N;
    float* wsum = ws + (size_t)2 * N;
    float* aggX = ws + (size_t)3 * N;
    float* acc  = ws + (size_t)3 * N + (size_t)N * SLOTS;  // 64x16 padded

    {
        size_t tot = (size_t)N * SLOTS;
        int blocks = (int)((tot + 255) / 256);
        gcn_init_ws<<<blocks, 256, 0, stream>>>(deg, wsum, aggX, acc, N);
    }
    gcn_degree<<<(E + 255) / 256, 256, 0, stream>>>(dstI, deg, E);
    gcn_dinv<<<(N + 255) / 256, 256, 0, stream>>>(deg, dinv, N);
    {
        int wavesPerBlock = 8;  // 256 threads = 8 wave32
        int blocks = (E + wavesPerBlock - 1) / wavesPerBlock;
        gcn_edge<<<blocks, 256, 0, stream>>>(srcI, dstI, x, dinv, wsum, aggX, E);
    }
    {
        int blocks = (N + NPB - 1) / NPB;
        gcn_node_reduce<<<blocks, 800, 0, stream>>>(x, dinv, wsum, aggX, W1, b1, acc, N);
    }
    gcn_final<<<1, 32, 0, stream>>>(acc, W2, b2, Wfc, bfc, (float*)d_out, 1.0f / (float)N);
}